// GAT_77678778516041
// MI455X (gfx1250) — compile-verified
//
#include <hip/hip_runtime.h>

// GAT propagate_attention for MI455X (gfx1250, wave32).
// Edge dot-products computed with V_WMMA_F32_16X16X4_F32 (16 edges/wave,
// diagonal of 16x16 K·Q^T), scatters via non-returning global f32 atomics.

typedef __attribute__((ext_vector_type(2))) float v2f;
typedef __attribute__((ext_vector_type(8))) float v8f;

#define N_NODES 50000
#define N_EDGES 800000
#define NH      4
#define DKDIM   112

// ---------------------------------------------------------------- zero d_out
__global__ void gat_zero_kernel(float* __restrict__ p, int n) {
  int i = blockIdx.x * blockDim.x + threadIdx.x;
  int stride = gridDim.x * blockDim.x;
  for (; i < n; i += stride) p[i] = 0.0f;
}

// ------------------------------------------------- pass 1: scores, z, wv
// One wave32 per 16-edge tile; 8 waves per 256-thread block.
// A (16x4, M=edge, K=feat): lane l holds M=l&15, K = 2*(l>>4)+{0,1}  (2 f32)
// B (4x16, N=edge)       : lane l holds N=l&15, K = 2*(l>>4)+{0,1}  (2 f32)
// C (16x16 f32)          : vgpr r, lanes 0-15 -> M=r,N=lane; lanes 16-31 -> M=r+8,N=lane-16
__global__ __launch_bounds__(256)
void gat_edge_kernel(const float* __restrict__ k,
                     const float* __restrict__ q,
                     const float* __restrict__ v,
                     const int*   __restrict__ src,
                     const int*   __restrict__ dst,
                     float* __restrict__ wv,
                     float* __restrict__ z,
                     float* __restrict__ score_ws)
{
  const int lane  = threadIdx.x & 31;
  const int wave  = threadIdx.x >> 5;
  const int tile  = (blockIdx.x * 8 + wave) * 16;      // first edge of tile
  const int eloc  = lane & 15;                         // my edge within tile
  const int kbase = (lane >> 4) * 2;                   // 0 or 2

  const int sv = src[tile + eloc];                     // src node of my edge
  const int dv = dst[tile + eloc];                     // dst node of my edge

  const float inv_sqrt_dk = 0.09449111825230679f;      // 1/sqrt(112)

  for (int h = 0; h < NH; ++h) {
    const float* __restrict__ krow = k + ((size_t)sv * NH + h) * DKDIM;
    const float* __restrict__ qrow = q + ((size_t)dv * NH + h) * DKDIM;

    v8f c = {0.f, 0.f, 0.f, 0.f, 0.f, 0.f, 0.f, 0.f};
#pragma unroll
    for (int k0 = 0; k0 < DKDIM; k0 += 4) {
      v2f a = *(const v2f*)(krow + k0 + kbase);        // global_load_b64
      v2f b = *(const v2f*)(qrow + k0 + kbase);
      // (neg_a, A, neg_b, B, c_mod, C, reuse_a, reuse_b)
      c = __builtin_amdgcn_wmma_f32_16x16x4_f32(
            false, a, false, b, (short)0, c, false, false);
    }

    // Extract diagonal element m (= raw dot for edge m) into lane m (m<16).
    // m<8 : c[m]   @ lane m ; m>=8 : c[m-8] @ lane m+16.
    float raw = 0.0f;
#pragma unroll
    for (int j = 0; j < 8; ++j) {
      float x = __shfl(c[j], (lane < 8) ? lane : lane + 16);
      if ((lane < 16) && (j == (lane & 7))) raw = x;
    }

    float score = __expf(fminf(fmaxf(raw * inv_sqrt_dk, -5.0f), 5.0f));

    if (lane < 16) {
      score_ws[(size_t)(tile + eloc) * NH + h] = score;
      atomicAdd(&z[(size_t)dv * NH + h], score);
    }

    // wv[dst] += score * v[src]  (whole wave per edge, 112 dims)
#pragma unroll 4
    for (int e = 0; e < 16; ++e) {
      float s  = __shfl(score, e);
      int   se = __shfl(sv, e);
      int   de = __shfl(dv, e);
      const float* __restrict__ vrow = v  + ((size_t)se * NH + h) * DKDIM;
      float*       __restrict__ wrow = wv + ((size_t)de * NH + h) * DKDIM;
      for (int d = lane; d < DKDIM; d += 32)
        atomicAdd(&wrow[d], s * vrow[d]);
    }
  }
}

// ------------------------------------------------- pass 2: new_flow
// new_flow[src[e],h] += flow_score[dst[e]] * score[e,h] / z[dst[e],h]
__global__ __launch_bounds__(256)
void gat_flow_kernel(const int*   __restrict__ src,
                     const int*   __restrict__ dst,
                     const float* __restrict__ flow_score,
                     const float* __restrict__ score_ws,
                     const float* __restrict__ z,
                     float* __restrict__ new_flow)
{
  int t = blockIdx.x * blockDim.x + threadIdx.x;
  if (t >= N_EDGES * NH) return;
  int e = t >> 2;           // NH == 4
  int h = t & 3;
  int s = src[e];
  int d = dst[e];
  float sc = score_ws[(size_t)e * NH + h];
  float zz = z[(size_t)d * NH + h];
  float val = flow_score[d] * sc / zz;
  atomicAdd(&new_flow[(size_t)s * NH + h], val);
}

// ---------------------------------------------------------------- launch
extern "C" void kernel_launch(void* const* d_in, const int* in_sizes, int n_in,
                              void* d_out, int out_size, void* d_ws, size_t ws_size,
                              hipStream_t stream) {
  const float* k    = (const float*)d_in[0];
  const float* q    = (const float*)d_in[1];
  const float* v    = (const float*)d_in[2];
  const float* flow = (const float*)d_in[3];
  const int*   src  = (const int*)d_in[4];
  const int*   dst  = (const int*)d_in[5];

  float* out = (float*)d_out;
  float* wv  = out;                                  // [N,H,DK]
  float* z   = out + (size_t)N_NODES * NH * DKDIM;   // [N,H]
  float* nf  = z   + (size_t)N_NODES * NH;           // [N,H]
  float* score_ws = (float*)d_ws;                    // [E,H] = 12.8 MB

  const int total_out = N_NODES * NH * DKDIM + 2 * N_NODES * NH;
  gat_zero_kernel<<<2048, 256, 0, stream>>>(out, total_out);

  // E = 800000 = 6250 blocks * 8 waves * 16 edges, exact (no tail)
  gat_edge_kernel<<<N_EDGES / (16 * 8), 256, 0, stream>>>(
      k, q, v, src, dst, wv, z, score_ws);

  gat_flow_kernel<<<(N_EDGES * NH + 255) / 256, 256, 0, stream>>>(
      src, dst, flow, score_ws, z, nf);
}